// SoftMoE_33517924778372
// MI455X (gfx1250) — compile-verified
//
#include <hip/hip_runtime.h>
#include <math.h>

// ---------------------------------------------------------------------------
// Soft-MoE forward for MI455X (gfx1250, wave32).
// - All GEMMs use V_WMMA_F32_16X16X4_F32 (fp32 end-to-end; pipeline is HBM
//   bandwidth bound, so fp32 matrix ops cost nothing vs bf16 and keep full
//   reference precision).
// - Row-major LDS panel staging uses GLOBAL_LOAD_ASYNC_TO_LDS_B128 (ASYNCcnt)
//   when the toolchain exposes the builtin; softmax-on-the-fly loaders stage
//   through VALU + ds_store.
// - dispatch/combine softmaxes are folded into GEMM A-loaders via per-row /
//   per-column (max, 1/sum) stats: 512 MB of intermediates never materialized.
// ---------------------------------------------------------------------------

typedef __attribute__((ext_vector_type(2))) float v2f;
typedef __attribute__((ext_vector_type(8))) float v8f;
typedef __attribute__((ext_vector_type(4))) int   v4i;

#define B_SZ   4
#define N_SEQ  4096
#define D_DIM  1024
#define E_EXP  8
#define S_SLOT 512
#define H_DIM  4096
#define ES     (E_EXP * S_SLOT)   // 4096
#define ROWS_X (B_SZ * N_SEQ)     // 16384
#define BS     (B_SZ * S_SLOT)    // 2048 rows per expert

// GEMM tiling: block = 256 threads = 8 waves (4 along M x 2 along N).
// Block tile 64 x 64; each wave owns a 16x32 C tile (2 accumulators, shared
// A fragment). K staged in panels of 32.
#define BM  64
#define BN  64
#define KC  32
#define LDA 36     // A panel stride: 144B rows (16B aligned), conflict-free frags
#define LDB 68     // B panel stride: 272B rows (16B aligned)

// ---------------------------------------------------------------------------
// Async global->LDS staging (gfx1250 GLOBAL_LOAD_ASYNC_TO_LDS_B128).
// Builtin signature wants typed v4i pointers: AS(1) src, AS(3) dst.
// ---------------------------------------------------------------------------
#if defined(__has_builtin)
#if __has_builtin(__builtin_amdgcn_global_load_async_to_lds_b128) && \
    __has_builtin(__builtin_amdgcn_s_wait_asynccnt)
#define HAS_ASYNC_LDS 1
#endif
#endif
#ifndef HAS_ASYNC_LDS
#define HAS_ASYNC_LDS 0
#endif

typedef __attribute__((address_space(1))) v4i* gv4i_ptr;
typedef __attribute__((address_space(3))) v4i* lv4i_ptr;

__device__ __forceinline__ void async_cp16(float* lds_dst, const float* gsrc) {
#if HAS_ASYNC_LDS
    __builtin_amdgcn_global_load_async_to_lds_b128(
        (gv4i_ptr)(gsrc), (lv4i_ptr)(lds_dst), 0, 0);
#else
    *(float4*)lds_dst = *(const float4*)gsrc;
#endif
}

__device__ __forceinline__ void async_wait() {
#if HAS_ASYNC_LDS
    __builtin_amdgcn_s_wait_asynccnt(0);
#endif
}

// ---------------------------------------------------------------------------
// WMMA inner loop over one KC-panel: one A fragment feeds two B tiles.
// A (16x4) frag: v0 = A[lr][kk+2*half], v1 = A[lr][kk+2*half+1]
// B (4x16) frag: v0 = B[kk+2*half][col], v1 = B[kk+2*half+1][col]
// la pre-offset to wave's 16 rows; lb pre-offset to wave's 32 cols.
// ---------------------------------------------------------------------------
__device__ __forceinline__ void wmma_kc2(const float* __restrict__ la,
                                         const float* __restrict__ lb,
                                         v8f& acc0, v8f& acc1) {
    const int lane = threadIdx.x & 31;
    const int half = lane >> 4;
    const int lr   = lane & 15;
#pragma unroll
    for (int kk = 0; kk < KC; kk += 4) {
        v2f av, bv0, bv1;
        av.x = la[lr * LDA + kk + 2 * half];
        av.y = la[lr * LDA + kk + 2 * half + 1];
        const float* b0 = lb + (kk + 2 * half) * LDB;
        const float* b1 = lb + (kk + 2 * half + 1) * LDB;
        bv0.x = b0[lr];      bv0.y = b1[lr];
        bv1.x = b0[lr + 16]; bv1.y = b1[lr + 16];
        acc0 = __builtin_amdgcn_wmma_f32_16x16x4_f32(false, av, false, bv0,
                                                     (short)0, acc0, false, false);
        acc1 = __builtin_amdgcn_wmma_f32_16x16x4_f32(false, av, false, bv1,
                                                     (short)0, acc1, false, false);
    }
}

__device__ __forceinline__ void tile_coords(int tid, int& waveM, int& waveN,
                                            int& half, int& lr) {
    const int wave = tid >> 5;
    waveM = wave & 3;       // 4 waves along M -> 64 rows
    waveN = wave >> 2;      // 2 waves along N -> 2 x 32 cols
    const int lane = tid & 31;
    half = lane >> 4;
    lr   = lane & 15;
}

// Async stage of a BM x KC row-major A panel (lda = row stride in floats).
__device__ __forceinline__ void stage_a_async(float* lds_a,
                                              const float* __restrict__ A,
                                              int m0, int k0, int lda, int tid) {
#pragma unroll
    for (int p = 0; p < 2; ++p) {
        const int chunk = tid + p * 256;          // 512 chunks of 4 floats
        const int i   = chunk >> 3;               // 8 chunks per 32-float row
        const int kk4 = (chunk & 7) * 4;
        async_cp16(&lds_a[i * LDA + kk4],
                   &A[(size_t)(m0 + i) * lda + k0 + kk4]);
    }
}

// Async stage of a KC x BN row-major B panel (ldb = row stride in floats).
__device__ __forceinline__ void stage_b_async(float* lds_b,
                                              const float* __restrict__ Bm,
                                              int k0, int n0, int ldb, int tid) {
#pragma unroll
    for (int p = 0; p < 2; ++p) {
        const int chunk = tid + p * 256;          // 512 chunks of 4 floats
        const int kk = chunk >> 4;                // 16 chunks per 64-float row
        const int j4 = (chunk & 15) * 4;
        async_cp16(&lds_b[kk * LDB + j4],
                   &Bm[(size_t)(k0 + kk) * ldb + n0 + j4]);
    }
}

// ---------------------------------------------------------------------------
// RMSNorm: one block per row of length d (=1024).
// y = x / max(||x||, 1e-12) * sqrt(d) * gamma
// ---------------------------------------------------------------------------
__global__ __launch_bounds__(256) void rmsnorm_kernel(
    const float* __restrict__ x, const float* __restrict__ g,
    float* __restrict__ y, int d) {
    __shared__ float red[8];
    __shared__ float snorm;
    const int row = blockIdx.x;
    const int tid = threadIdx.x;
    const float* xr = x + (size_t)row * d;
    float ss = 0.f;
    for (int i = tid; i < d; i += 256) { float v = xr[i]; ss += v * v; }
    for (int off = 16; off; off >>= 1) ss += __shfl_xor(ss, off, 32);
    if ((tid & 31) == 0) red[tid >> 5] = ss;
    __syncthreads();
    if (tid == 0) {
        float t = 0.f;
        for (int i = 0; i < 8; ++i) t += red[i];
        snorm = sqrtf((float)d) / fmaxf(sqrtf(t), 1e-12f);
    }
    __syncthreads();
    const float sc = snorm;
    float* yr = y + (size_t)row * d;
    for (int i = tid; i < d; i += 256) yr[i] = xr[i] * sc * g[i];
}

// ---------------------------------------------------------------------------
// GEMM 1: logits[row, es] = xn[row, :] . se[es, :]   (K = 1024)
// A async-staged; B needs a transpose so it stages through VALU.
// ---------------------------------------------------------------------------
__global__ __launch_bounds__(256) void gemm_logits(
    const float* __restrict__ xn, const float* __restrict__ se,
    float* __restrict__ logits) {
    __shared__ float lds_a[BM * LDA];
    __shared__ float lds_b[KC * LDB];
    const int tid = threadIdx.x;
    const int m0 = blockIdx.x * BM;
    const int n0 = blockIdx.y * BN;
    int waveM, waveN, half, lr;
    tile_coords(tid, waveM, waveN, half, lr);
    v8f acc0 = {}, acc1 = {};
    for (int k0 = 0; k0 < D_DIM; k0 += KC) {
        stage_a_async(lds_a, xn, m0, k0, D_DIM, tid);
        {   // B: se[n0+j][k0+kk] -> lds_b[kk][j]; contiguous in k
            const int kk = tid & 31, j0 = tid >> 5;
#pragma unroll
            for (int p = 0; p < 8; ++p) {
                const int j = j0 + p * 8;
                lds_b[kk * LDB + j] = se[(size_t)(n0 + j) * D_DIM + k0 + kk];
            }
        }
        async_wait();
        __syncthreads();
        wmma_kc2(&lds_a[(waveM * 16) * LDA], &lds_b[waveN * 32], acc0, acc1);
        __syncthreads();
    }
    const int cm  = m0 + waveM * 16 + 8 * half;
    const int cn0 = n0 + waveN * 32 + lr;
#pragma unroll
    for (int r = 0; r < 8; ++r) {
        logits[(size_t)(cm + r) * ES + cn0]      = acc0[r];
        logits[(size_t)(cm + r) * ES + cn0 + 16] = acc1[r];
    }
}

// ---------------------------------------------------------------------------
// Column softmax stats (dispatch): per (b, es), online max/sum over n.
// ---------------------------------------------------------------------------
__global__ __launch_bounds__(256) void colstats_kernel(
    const float* __restrict__ logits, float* __restrict__ cmax,
    float* __restrict__ cinv) {
    const int idx = blockIdx.x * 256 + threadIdx.x;   // 0 .. B*ES-1
    const int b = idx / ES;
    const int es = idx - b * ES;
    const float* p = logits + (size_t)b * N_SEQ * ES + es;
    float m = -INFINITY, s = 0.f;
    for (int nI = 0; nI < N_SEQ; ++nI) {
        const float v = p[(size_t)nI * ES];
        if (v > m) { s *= expf(m - v); m = v; }
        s += expf(v - m);
    }
    cmax[idx] = m;
    cinv[idx] = 1.0f / s;
}

// ---------------------------------------------------------------------------
// Row softmax stats (combine): per (b,n) row, max/sum over es (contiguous).
// ---------------------------------------------------------------------------
__global__ __launch_bounds__(256) void rowstats_kernel(
    const float* __restrict__ logits, float* __restrict__ rmax,
    float* __restrict__ rinv) {
    __shared__ float red[8];
    __shared__ float Msh;
    const int row = blockIdx.x;
    const int tid = threadIdx.x;
    const float* p = logits + (size_t)row * ES;
    float m = -INFINITY;
    for (int i = tid; i < ES; i += 256) m = fmaxf(m, p[i]);
    for (int off = 16; off; off >>= 1) m = fmaxf(m, __shfl_xor(m, off, 32));
    if ((tid & 31) == 0) red[tid >> 5] = m;
    __syncthreads();
    if (tid == 0) {
        float t = -INFINITY;
        for (int i = 0; i < 8; ++i) t = fmaxf(t, red[i]);
        Msh = t;
    }
    __syncthreads();
    const float mm = Msh;
    float s = 0.f;
    for (int i = tid; i < ES; i += 256) s += expf(p[i] - mm);
    for (int off = 16; off; off >>= 1) s += __shfl_xor(s, off, 32);
    if ((tid & 31) == 0) red[tid >> 5] = s;
    __syncthreads();
    if (tid == 0) {
        float t = 0.f;
        for (int i = 0; i < 8; ++i) t += red[i];
        rmax[row] = mm;
        rinv[row] = 1.0f / t;
    }
}

// ---------------------------------------------------------------------------
// GEMM 2 (slots): per b, slots_es[m, d] = sum_n dispatch[b,n,m] * xn[b,n,d]
// A[m,k] = exp(logits[b,k,m]-cmax)*cinv applied on LDS stage (VALU path);
// B async-staged. K = n = 4096. Output scattered to [e, b, s, d].
// ---------------------------------------------------------------------------
__global__ __launch_bounds__(256) void gemm_slots(
    const float* __restrict__ logits, const float* __restrict__ xn,
    const float* __restrict__ cmax, const float* __restrict__ cinv,
    float* __restrict__ slots) {
    __shared__ float lds_a[BM * LDA];
    __shared__ float lds_b[KC * LDB];
    const int tid = threadIdx.x;
    const int b  = blockIdx.z;
    const int m0 = blockIdx.x * BM;   // es
    const int n0 = blockIdx.y * BN;   // d
    int waveM, waveN, half, lr;
    tile_coords(tid, waveM, waveN, half, lr);
    const int aj  = tid & 63;                      // fixed column m = m0+aj
    const int ak0 = tid >> 6;                      // 0..3
    const float aMax = cmax[b * ES + m0 + aj];
    const float aInv = cinv[b * ES + m0 + aj];
    v8f acc0 = {}, acc1 = {};
    for (int k0 = 0; k0 < N_SEQ; k0 += KC) {
        stage_b_async(lds_b, xn + (size_t)b * N_SEQ * D_DIM, k0, n0, D_DIM, tid);
        {   // A: dispatch-on-the-fly; global contiguous in m (=aj)
#pragma unroll
            for (int p = 0; p < 8; ++p) {
                const int kk = ak0 + p * 4;
                const float v =
                    logits[((size_t)b * N_SEQ + (k0 + kk)) * ES + m0 + aj];
                lds_a[aj * LDA + kk] = expf(v - aMax) * aInv;
            }
        }
        async_wait();
        __syncthreads();
        wmma_kc2(&lds_a[(waveM * 16) * LDA], &lds_b[waveN * 32], acc0, acc1);
        __syncthreads();
    }
    const int cm  = m0 + waveM * 16 + 8 * half;   // es index
    const int cn0 = n0 + waveN * 32 + lr;
#pragma unroll
    for (int r = 0; r < 8; ++r) {
        const int m = cm + r;
        const int srow = (m >> 9) * (B_SZ * S_SLOT) + b * S_SLOT + (m & 511);
        slots[(size_t)srow * D_DIM + cn0]      = acc0[r];
        slots[(size_t)srow * D_DIM + cn0 + 16] = acc1[r];
    }
}

// ---------------------------------------------------------------------------
// GEMM 3 (hid): per e, hid[m, h] = gelu(slots_e[m,:] . w1_e[:,h] + b1_e[h])
// Both panels async-staged. K = d = 1024, exact gelu.
// ---------------------------------------------------------------------------
__global__ __launch_bounds__(256) void gemm_hid(
    const float* __restrict__ slots, const float* __restrict__ w1,
    const float* __restrict__ b1, float* __restrict__ hid) {
    __shared__ float lds_a[BM * LDA];
    __shared__ float lds_b[KC * LDB];
    const int tid = threadIdx.x;
    const int e  = blockIdx.z;
    const int m0 = blockIdx.x * BM;   // bs rows
    const int n0 = blockIdx.y * BN;   // h
    const float* A  = slots + (size_t)e * BS * D_DIM;
    const float* Bm = w1 + (size_t)e * D_DIM * H_DIM;
    int waveM, waveN, half, lr;
    tile_coords(tid, waveM, waveN, half, lr);
    v8f acc0 = {}, acc1 = {};
    for (int k0 = 0; k0 < D_DIM; k0 += KC) {
        stage_a_async(lds_a, A, m0, k0, D_DIM, tid);
        stage_b_async(lds_b, Bm, k0, n0, H_DIM, tid);
        async_wait();
        __syncthreads();
        wmma_kc2(&lds_a[(waveM * 16) * LDA], &lds_b[waveN * 32], acc0, acc1);
        __syncthreads();
    }
    const int cm  = m0 + waveM * 16 + 8 * half;
    const int cn0 = n0 + waveN * 32 + lr;
    const float bias0 = b1[e * H_DIM + cn0];
    const float bias1 = b1[e * H_DIM + cn0 + 16];
    float* O = hid + (size_t)e * BS * H_DIM;
#pragma unroll
    for (int r = 0; r < 8; ++r) {
        const float h0 = acc0[r] + bias0;
        const float h1 = acc1[r] + bias1;
        O[(size_t)(cm + r) * H_DIM + cn0] =
            0.5f * h0 * (1.0f + erff(h0 * 0.70710678118654752f));
        O[(size_t)(cm + r) * H_DIM + cn0 + 16] =
            0.5f * h1 * (1.0f + erff(h1 * 0.70710678118654752f));
    }
}

// ---------------------------------------------------------------------------
// GEMM 4 (out): per e, outs[m, d] = hid_e[m,:] . w2_e[:,d] + b2_e[d]
// Both panels async-staged. K = h = 4096.
// ---------------------------------------------------------------------------
__global__ __launch_bounds__(256) void gemm_out(
    const float* __restrict__ hid, const float* __restrict__ w2,
    const float* __restrict__ b2, float* __restrict__ outs) {
    __shared__ float lds_a[BM * LDA];
    __shared__ float lds_b[KC * LDB];
    const int tid = threadIdx.x;
    const int e  = blockIdx.z;
    const int m0 = blockIdx.x * BM;
    const int n0 = blockIdx.y * BN;
    const float* A  = hid + (size_t)e * BS * H_DIM;
    const float* Bm = w2 + (size_t)e * H_DIM * D_DIM;
    int waveM, waveN, half, lr;
    tile_coords(tid, waveM, waveN, half, lr);
    v8f acc0 = {}, acc1 = {};
    for (int k0 = 0; k0 < H_DIM; k0 += KC) {
        stage_a_async(lds_a, A, m0, k0, H_DIM, tid);
        stage_b_async(lds_b, Bm, k0, n0, D_DIM, tid);
        async_wait();
        __syncthreads();
        wmma_kc2(&lds_a[(waveM * 16) * LDA], &lds_b[waveN * 32], acc0, acc1);
        __syncthreads();
    }
    const int cm  = m0 + waveM * 16 + 8 * half;
    const int cn0 = n0 + waveN * 32 + lr;
    const float bias0 = b2[e * D_DIM + cn0];
    const float bias1 = b2[e * D_DIM + cn0 + 16];
    float* O = outs + (size_t)e * BS * D_DIM;
#pragma unroll
    for (int r = 0; r < 8; ++r) {
        O[(size_t)(cm + r) * D_DIM + cn0]      = acc0[r] + bias0;
        O[(size_t)(cm + r) * D_DIM + cn0 + 16] = acc1[r] + bias1;
    }
}

// ---------------------------------------------------------------------------
// GEMM 5 (final): per b, out[n, d] = sum_es combine[b,n,es] * out_t[b,es,d]
// A[m,k] = exp(logits[b,m,k]-rmax)*rinv on the fly (VALU path); B rows
// gathered from [e, b, s, d] layout, async-staged. K = es = 4096.
// ---------------------------------------------------------------------------
__global__ __launch_bounds__(256) void gemm_final(
    const float* __restrict__ logits, const float* __restrict__ outs,
    const float* __restrict__ rmax, const float* __restrict__ rinv,
    float* __restrict__ out) {
    __shared__ float lds_a[BM * LDA];
    __shared__ float lds_b[KC * LDB];
    const int tid = threadIdx.x;
    const int b  = blockIdx.z;
    const int m0 = blockIdx.x * BM;   // n
    const int n0 = blockIdx.y * BN;   // d
    int waveM, waveN, half, lr;
    tile_coords(tid, waveM, waveN, half, lr);
    v8f acc0 = {}, acc1 = {};
    for (int k0 = 0; k0 < ES; k0 += KC) {
        {   // B: gather rows from [e, b, s, d]; 16B chunks, async
#pragma unroll
            for (int p = 0; p < 2; ++p) {
                const int chunk = tid + p * 256;
                const int kk = chunk >> 4;
                const int j4 = (chunk & 15) * 4;
                const int kg = k0 + kk;
                const int brow =
                    (kg >> 9) * (B_SZ * S_SLOT) + b * S_SLOT + (kg & 511);
                async_cp16(&lds_b[kk * LDB + j4],
                           &outs[(size_t)brow * D_DIM + n0 + j4]);
            }
        }
        {   // A: combine-on-the-fly; logits row-major contiguous in k
            const int kk = tid & 31, i0 = tid >> 5;
#pragma unroll
            for (int p = 0; p < 8; ++p) {
                const int i = i0 + p * 8;
                const int row = b * N_SEQ + m0 + i;
                const float v = logits[(size_t)row * ES + k0 + kk];
                lds_a[i * LDA + kk] = expf(v - rmax[row]) * rinv[row];
            }
        }
        async_wait();
        __syncthreads();
        wmma_kc2(&lds_a[(waveM * 16) * LDA], &lds_b[waveN * 32], acc0, acc1);
        __syncthreads();
    }
    const int cm  = m0 + waveM * 16 + 8 * half;
    const int cn0 = n0 + waveN * 32 + lr;
#pragma unroll
    for (int r = 0; r < 8; ++r) {
        out[((size_t)b * N_SEQ + cm + r) * D_DIM + cn0]      = acc0[r];
        out[((size_t)b * N_SEQ + cm + r) * D_DIM + cn0 + 16] = acc1[r];
    }
}

// ---------------------------------------------------------------------------
// Launch: inputs in setup_inputs() order:
// 0:x 1:gamma 2:slot_gamma 3:slot_embeds 4:w1 5:b1 6:w2 7:b2
// ---------------------------------------------------------------------------
extern "C" void kernel_launch(void* const* d_in, const int* in_sizes, int n_in,
                              void* d_out, int out_size, void* d_ws,
                              size_t ws_size, hipStream_t stream) {
    const float* x          = (const float*)d_in[0];
    const float* gamma      = (const float*)d_in[1];
    const float* slot_gamma = (const float*)d_in[2];
    const float* slot_emb   = (const float*)d_in[3];
    const float* w1         = (const float*)d_in[4];
    const float* b1         = (const float*)d_in[5];
    const float* w2         = (const float*)d_in[6];
    const float* b2         = (const float*)d_in[7];
    float* out = (float*)d_out;
    float* ws  = (float*)d_ws;

    // Workspace layout (floats); total ~188.8M floats (~755 MB)
    float* xn     = ws;                                   // 16384*1024
    float* se     = xn + (size_t)ROWS_X * D_DIM;          // 4096*1024
    float* logits = se + (size_t)ES * D_DIM;              // 16384*4096
    float* cmax   = logits + (size_t)ROWS_X * ES;         // 16384
    float* cinv   = cmax + (size_t)B_SZ * ES;             // 16384
    float* rmaxv  = cinv + (size_t)B_SZ * ES;             // 16384
    float* rinvv  = rmaxv + (size_t)ROWS_X;               // 16384
    float* slots  = rinvv + (size_t)ROWS_X;               // 8*2048*1024
    float* hid    = slots + (size_t)E_EXP * BS * D_DIM;   // 8*2048*4096
    float* outs   = hid + (size_t)E_EXP * BS * H_DIM;     // 8*2048*1024

    // 1) RMSNorm of x and slot_embeds
    rmsnorm_kernel<<<ROWS_X, 256, 0, stream>>>(x, gamma, xn, D_DIM);
    rmsnorm_kernel<<<ES, 256, 0, stream>>>(slot_emb, slot_gamma, se, D_DIM);

    // 2) logits = xn . se^T
    gemm_logits<<<dim3(ROWS_X / BM, ES / BN), 256, 0, stream>>>(xn, se, logits);

    // 3) softmax stats
    colstats_kernel<<<(B_SZ * ES) / 256, 256, 0, stream>>>(logits, cmax, cinv);
    rowstats_kernel<<<ROWS_X, 256, 0, stream>>>(logits, rmaxv, rinvv);

    // 4) slots = dispatch^T . xn   (per b)
    gemm_slots<<<dim3(ES / BM, D_DIM / BN, B_SZ), 256, 0, stream>>>(
        logits, xn, cmax, cinv, slots);

    // 5) hid = gelu(slots . w1 + b1)   (per e)
    gemm_hid<<<dim3(BS / BM, H_DIM / BN, E_EXP), 256, 0, stream>>>(
        slots, w1, b1, hid);

    // 6) outs = hid . w2 + b2   (per e)
    gemm_out<<<dim3(BS / BM, D_DIM / BN, E_EXP), 256, 0, stream>>>(
        hid, w2, b2, outs);

    // 7) out = combine . out_t   (per b)
    gemm_final<<<dim3(N_SEQ / BM, D_DIM / BN, B_SZ), 256, 0, stream>>>(
        logits, outs, rmaxv, rinvv, out);
}